// DirectedGCNConv_40776419508522
// MI455X (gfx1250) — compile-verified
//
#include <hip/hip_runtime.h>
#include <hip/hip_bf16.h>

typedef __attribute__((ext_vector_type(2))) float v2f;
typedef __attribute__((ext_vector_type(8))) float v8f;

#define D 128            // D_IN == D_OUT == 128
#define D4 (D / 4)       // float4s per row
#define ROWS 64          // rows per block (4 row-tiles of 16)

// ---------------------------------------------------------------------------
// Zero the accumulation buffers (agg: n_nodes*D floats, indeg: n_nodes floats,
// laid out contiguously). Must run every call (graph replay, no re-poison).
// ---------------------------------------------------------------------------
__global__ void zero_kernel(float* __restrict__ p, long long n) {
    long long i = (long long)blockIdx.x * blockDim.x + threadIdx.x;
    if (i < n) p[i] = 0.0f;
}

// ---------------------------------------------------------------------------
// h = x @ weight using V_WMMA_F32_16X16X4_F32 (fp32 in / fp32 acc, wave32).
// Block = 256 threads = 8 waves; block computes a 64-row slab of h.
// Wave w owns columns [16w, 16w+16) and 4 row-tiles -> 4 accumulators,
// 128 WMMAs/wave. B fragment loaded once per K-step and reused across the
// 4 row-tiles (4x fewer weight loads than one-tile-per-block; 4 independent
// WMMAs back-to-back keep the matrix pipe busy without RAW hazards).
//
// Fragment layout (ISA 7.12.2, wave32):
//   A (16x4 f32, 2 VGPRs): lanes 0-15 row=lane, K={kb,kb+1};
//                          lanes 16-31 row=lane-16, K={kb+2,kb+3}
//   B (4x16 f32, 2 VGPRs): VGPR v: lanes 0-15 row K=kb+v, lanes 16-31 K=kb+v+2
//   C/D (16x16 f32, 8 VGPRs): VGPR v: lanes 0-15 row v, lanes 16-31 row v+8
// ---------------------------------------------------------------------------
__global__ __launch_bounds__(256) void gemm_wmma_kernel(
    const float* __restrict__ x, const float* __restrict__ w,
    float* __restrict__ h, int n_nodes) {
    __shared__ float As[ROWS * D];               // 32 KB x-slab

    const long long row0 = (long long)blockIdx.x * ROWS;

    // Cooperative staged load of the 64xD x-slab into LDS (float4 / b128).
    for (int i = threadIdx.x; i < ROWS * D4; i += 256) {
        const int r = i / D4;
        float4 v = make_float4(0.f, 0.f, 0.f, 0.f);
        if (row0 + r < n_nodes)
            v = reinterpret_cast<const float4*>(x)[(row0 + r) * D4 + (i % D4)];
        reinterpret_cast<float4*>(As)[i] = v;
    }
    __syncthreads();

    const int wave = threadIdx.x >> 5;
    const int lane = threadIdx.x & 31;
    const int half = lane >> 4;                  // 0: lanes 0-15, 1: lanes 16-31
    const int m    = lane & 15;                  // A row / B,C column within tile
    const int n0   = wave * 16;                  // this wave's column tile

    v8f c0 = {}, c1 = {}, c2 = {}, c3 = {};
    #pragma unroll
    for (int kb = 0; kb < D; kb += 4) {
        const int k0 = kb + 2 * half;
        v2f b;                                   // shared across the 4 row-tiles
        b.x = w[(long long)k0 * D + n0 + m];
        b.y = w[(long long)(k0 + 1) * D + n0 + m];
        v2f a0, a1, a2, a3;                      // contiguous pairs -> ds_load_b64
        a0.x = As[(m     ) * D + k0]; a0.y = As[(m     ) * D + k0 + 1];
        a1.x = As[(m + 16) * D + k0]; a1.y = As[(m + 16) * D + k0 + 1];
        a2.x = As[(m + 32) * D + k0]; a2.y = As[(m + 32) * D + k0 + 1];
        a3.x = As[(m + 48) * D + k0]; a3.y = As[(m + 48) * D + k0 + 1];
        c0 = __builtin_amdgcn_wmma_f32_16x16x4_f32(false, a0, false, b, (short)0, c0, false, false);
        c1 = __builtin_amdgcn_wmma_f32_16x16x4_f32(false, a1, false, b, (short)0, c1, false, false);
        c2 = __builtin_amdgcn_wmma_f32_16x16x4_f32(false, a2, false, b, (short)0, c2, false, false);
        c3 = __builtin_amdgcn_wmma_f32_16x16x4_f32(false, a3, false, b, (short)0, c3, false, false);
    }

    // Write back the four 16x16 tiles.
    #pragma unroll
    for (int v = 0; v < 8; ++v) {
        const long long rr = row0 + v + 8 * half;
        if (rr      < n_nodes) h[(rr     ) * D + n0 + m] = c0[v];
        if (rr + 16 < n_nodes) h[(rr + 16) * D + n0 + m] = c1[v];
        if (rr + 32 < n_nodes) h[(rr + 32) * D + n0 + m] = c2[v];
        if (rr + 48 < n_nodes) h[(rr + 48) * D + n0 + m] = c3[v];
    }
}

// ---------------------------------------------------------------------------
// Edge phase: one wave per edge. float4 gather of h[src], scale by edge
// weight, 4 fp32 atomic adds per lane into agg[tgt]. Lane 0 also accumulates
// the weighted in-degree. h+agg (~102 MB) fit in the 192 MB L2, so the
// random gathers and the 82M fp32 atomics stay L2-resident.
// ---------------------------------------------------------------------------
__global__ __launch_bounds__(256) void scatter_kernel(
    const float* __restrict__ h, const int* __restrict__ edge_index,
    const float* __restrict__ ew, float* __restrict__ agg,
    float* __restrict__ indeg, int n_edges) {
    const int e = blockIdx.x * (blockDim.x >> 5) + (threadIdx.x >> 5);
    if (e >= n_edges) return;
    const int lane = threadIdx.x & 31;

    const int   src = edge_index[e];             // edge_index[0][e]
    const int   tgt = edge_index[n_edges + e];   // edge_index[1][e]
    const float wgt = ew[e];

    const float4 v = reinterpret_cast<const float4*>(h + (long long)src * D)[lane];
    float* at = agg + (long long)tgt * D + 4 * lane;
    atomicAdd(at + 0, v.x * wgt);
    atomicAdd(at + 1, v.y * wgt);
    atomicAdd(at + 2, v.z * wgt);
    atomicAdd(at + 3, v.w * wgt);
    if (lane == 0) atomicAdd(indeg + tgt, wgt);
}

// ---------------------------------------------------------------------------
// out = (h + agg) * rsqrt(max(indeg,1)) + bias, vectorized float4.
// ---------------------------------------------------------------------------
__global__ __launch_bounds__(256) void finalize_kernel(
    const float* __restrict__ h, const float* __restrict__ agg,
    const float* __restrict__ indeg, const float* __restrict__ bias,
    float* __restrict__ out, int n_nodes) {
    const long long i = (long long)blockIdx.x * blockDim.x + threadIdx.x;
    const long long total = (long long)n_nodes * D4;
    if (i >= total) return;
    const int node = (int)(i >> 5);              // D4 == 32 float4s per row
    const int q    = (int)(i & 31);

    const float  dinv = rsqrtf(fmaxf(indeg[node], 1.0f));
    const float4 hv = reinterpret_cast<const float4*>(h)[i];
    const float4 av = reinterpret_cast<const float4*>(agg)[i];
    const float4 bv = reinterpret_cast<const float4*>(bias)[q];

    float4 o;
    o.x = (hv.x + av.x) * dinv + bv.x;
    o.y = (hv.y + av.y) * dinv + bv.y;
    o.z = (hv.z + av.z) * dinv + bv.z;
    o.w = (hv.w + av.w) * dinv + bv.w;
    reinterpret_cast<float4*>(out)[i] = o;
}

// ---------------------------------------------------------------------------
// Workspace layout (floats): [agg: n*D][indeg: n][h: n*D]  (~103 MB)
// ---------------------------------------------------------------------------
extern "C" void kernel_launch(void* const* d_in, const int* in_sizes, int n_in,
                              void* d_out, int out_size, void* d_ws, size_t ws_size,
                              hipStream_t stream) {
    const float* x    = (const float*)d_in[0];
    const int*   ei   = (const int*)d_in[1];
    const float* ew   = (const float*)d_in[2];
    const float* w    = (const float*)d_in[3];
    const float* bias = (const float*)d_in[4];
    float*       out  = (float*)d_out;

    const int n_nodes = in_sizes[0] / D;
    const int n_edges = in_sizes[2];

    float* agg   = (float*)d_ws;
    float* indeg = agg + (long long)n_nodes * D;
    float* h     = indeg + n_nodes;

    // 1. Zero agg + indeg (contiguous: n_nodes*D + n_nodes floats).
    {
        const long long nz = (long long)n_nodes * D + n_nodes;
        const int blocks = (int)((nz + 255) / 256);
        zero_kernel<<<blocks, 256, 0, stream>>>(agg, nz);
    }
    // 2. h = x @ weight (WMMA, 64-row slabs).
    {
        const int blocks = (n_nodes + ROWS - 1) / ROWS;
        gemm_wmma_kernel<<<blocks, 256, 0, stream>>>(x, w, h, n_nodes);
    }
    // 3. Edge gather/scale/scatter + in-degree. 8 edges (waves) per block.
    {
        const int blocks = (n_edges + 7) / 8;
        scatter_kernel<<<blocks, 256, 0, stream>>>(h, ei, ew, agg, indeg, n_edges);
    }
    // 4. Fused normalization + bias epilogue.
    {
        const long long total = (long long)n_nodes * D4;
        const int blocks = (int)((total + 255) / 256);
        finalize_kernel<<<blocks, 256, 0, stream>>>(h, agg, indeg, bias, out, n_nodes);
    }
}